// HawkesBase_14053132992795
// MI455X (gfx1250) — compile-verified
//
#include <hip/hip_runtime.h>

// ---------------- problem constants ----------------
#define MM   50          // marks
#define KK   4           // kernels
#define MK   200         // MM*KK state size
#define CPAD 256         // contraction columns seen by WMMA (200 real + 56 zero)
#define RS   260         // LDS row stride in floats (260 % 64 == 4 -> conflict-free)
#define CB   512         // events per block (chunk)
#define TILE 16          // WMMA tile of events
#define NT   256         // threads per block (8 wave32)
#define NW   8           // waves per block
#define KPW  8           // WMMA K-steps per wave (CPAD/4/NW)

typedef __attribute__((ext_vector_type(2))) float v2f;
typedef __attribute__((ext_vector_type(8))) float v8f;

// ============================================================
// Kernel 0: pre-scaled transposed alpha table.
//   alphaT[mn*200 + p] = gamma_k * alpha[k, mn, m],  p = k*50 + m
// ============================================================
__global__ void hawkes_k0_alphaT(const float* __restrict__ alpha,
                                 const float* __restrict__ gamma,
                                 float* __restrict__ alphaT) {
    const int mn = blockIdx.x;           // 0..MM-1
    const int p  = threadIdx.x;
    if (p < MK) {
        const int k = p / MM;
        const int m = p - k * MM;
        alphaT[mn * MK + p] = gamma[k] * alpha[k * (MM * MM) + mn * MM + m];
    }
}

// ============================================================
// Kernel 1: per-block local summary.
//   L_b[m,k] = sum_{j in block} [mi_j==m] * exp(-g_k*(t_next - t_j))
//   A_b[k]   = exp(-g_k*(t_next - t_blockstart))
// All exponents <= 0 -> values in (0,1], no overflow. Deterministic.
// ============================================================
__global__ void hawkes_k1_local(const float* __restrict__ ti,
                                const int* __restrict__ mi,
                                const float* __restrict__ gamma,
                                float* __restrict__ Lw,
                                float* __restrict__ Aw,
                                int N) {
    __shared__ float ti_sh[CB];
    __shared__ int   mi_sh[CB];
    __shared__ float gsh[KK];

    const int b   = blockIdx.x;
    const int s   = b * CB;
    const int cnt = min(CB, N - s);

    for (int j = threadIdx.x; j < cnt; j += NT) {
        ti_sh[j] = ti[s + j];
        mi_sh[j] = mi[s + j];
    }
    if (threadIdx.x < KK) gsh[threadIdx.x] = gamma[threadIdx.x];
    __syncthreads();

    const float t_next = (s + cnt < N) ? ti[s + cnt] : ti[N - 1];

    const int p = threadIdx.x;
    if (p < MK) {
        const int   k = p / MM;
        const int   m = p - k * MM;
        const float g = gsh[k];
        float acc = 0.f;
        for (int j = 0; j < cnt; ++j) {
            if (mi_sh[j] == m) acc += __expf(-g * (t_next - ti_sh[j]));
        }
        Lw[(size_t)b * MK + p] = acc;
        if (m == 0) Aw[(size_t)b * KK + k] = __expf(-g * (t_next - ti_sh[0]));
    }
}

// ============================================================
// Kernel 2: sequential carry scan across blocks (one workgroup).
//   H_0 = 0 ; store H_b ; H_{b+1} = A_b * H_b + L_b
// ============================================================
__global__ void hawkes_k2_scan(const float* __restrict__ Lw,
                               const float* __restrict__ Aw,
                               float* __restrict__ Hw,
                               int B) {
    const int p = threadIdx.x;
    if (p >= MK) return;
    const int k = p / MM;
    float H = 0.f;
#pragma unroll 4
    for (int b = 0; b < B; ++b) {
        if (b + 8 < B) {
            // gfx1250 global_prefetch_b8: pull future L rows into cache
            __builtin_prefetch(&Lw[(size_t)(b + 8) * MK + p], 0, 0);
        }
        Hw[(size_t)b * MK + p] = H;
        const float a = Aw[(size_t)b * KK + k];
        const float l = Lw[(size_t)b * MK + p];
        H = fmaf(a, H, l);
    }
}

// ============================================================
// Kernel 3: per block, replay the exact recursion with correct
// initial state H_b, in 16-event tiles.
//   phase 1a: X[i,p] = alphaT[mi_i][p]  (16 independent coalesced loads)
//   phase 1b: precompute 16 exps, then pure 16-FMA state chain; Y[i,p]=S
//   phase 2 : each wave runs a COMPILE-TIME 8-step WMMA slice of the
//             256-wide padded contraction (pad cols are zero)
//   phase 3 : lanes 0..15 reduce the 8 wave partials in fixed order
// ============================================================
__global__ void hawkes_k3_main(const float* __restrict__ ti,
                               const int* __restrict__ mi,
                               const float* __restrict__ mu,
                               const float* __restrict__ gamma,
                               const float* __restrict__ alphaT,
                               const float* __restrict__ Hw,
                               float* __restrict__ partial,
                               int N) {
    __shared__ float ti_sh[CB + 1];   // halo: ti[s-1]
    __shared__ int   mi_sh[CB + 1];   // halo: mi[s-1]
    __shared__ float mu_sh[CB];
    __shared__ float X_sh[TILE * RS];
    __shared__ float Y_sh[TILE * RS];
    __shared__ float Dpart[NW * TILE];
    __shared__ float acc_sh[TILE];
    __shared__ float gsh[KK];

    const int b   = blockIdx.x;
    const int s   = b * CB;
    const int cnt = min(CB, N - s);
    const int tid = threadIdx.x;

    for (int j = tid; j < cnt; j += NT) {
        ti_sh[j + 1] = ti[s + j];
        mi_sh[j + 1] = mi[s + j];
        mu_sh[j]     = mu[s + j];
    }
    if (tid == 0) {
        ti_sh[0] = (s > 0) ? ti[s - 1] : 0.f;
        mi_sh[0] = (s > 0) ? mi[s - 1] : 0;
    }
    if (tid < KK) gsh[tid] = gamma[tid];

    // zero the pad columns (200..255) once; never touched again
    if (tid >= MK) {
#pragma unroll
        for (int i = 0; i < TILE; ++i) {
            X_sh[i * RS + tid] = 0.f;
            Y_sh[i * RS + tid] = 0.f;
        }
    }
    __syncthreads();

    const int   p = tid;
    const int   k = (p < MK) ? (p / MM) : 0;
    const int   m = p - k * MM;
    const float g = (p < MK) ? gsh[k] : 0.f;
    float S   = (p < MK) ? Hw[(size_t)b * MK + p] : 0.f;  // state at first event
    float acc = 0.f;

    const int ntiles = (cnt + TILE - 1) / TILE;
    const int wv   = tid >> 5;          // wave id (wave-uniform)
    const int L    = tid & 31;
    const int row  = L & 15;
    const int koff = (L >> 4) << 1;     // 0 or 2

    for (int t = 0; t < ntiles; ++t) {
        // ---- phase 1a: X fill, 16 independent coalesced loads ----
        if (p < MK) {
            float xv[TILE];
#pragma unroll
            for (int i = 0; i < TILE; ++i) {
                const int li = t * TILE + i;
                xv[i] = (li < cnt) ? alphaT[mi_sh[li + 1] * MK + p] : 0.f;
            }
#pragma unroll
            for (int i = 0; i < TILE; ++i) X_sh[i * RS + p] = xv[i];

            // ---- phase 1b: precompute exps, then pure FMA chain ----
            float av[TILE], ov[TILE];
#pragma unroll
            for (int i = 0; i < TILE; ++i) {
                const int li = t * TILE + i;
                const int gn = s + li;
                const bool upd = (li < cnt) && (gn > 0);
                const float dt = upd ? (ti_sh[li + 1] - ti_sh[li]) : 0.f;
                av[i] = upd ? __expf(-g * dt) : 1.f;
                ov[i] = (upd && (mi_sh[li] == m)) ? 1.f : 0.f;
            }
#pragma unroll
            for (int i = 0; i < TILE; ++i) {
                S = av[i] * (S + ov[i]);
                Y_sh[i * RS + p] = S;
            }
        }
        __syncthreads();

        // ---- phase 2: per-wave fixed 8-step WMMA slice ----
        {
            const int base = row * RS + wv * (KPW * 4) + koff;
            const float* xb = &X_sh[base];
            const float* yb = &Y_sh[base];
            v8f c = {};
#pragma unroll
            for (int j = 0; j < KPW; ++j) {
                v2f avv, bvv;
                avv.x = xb[j * 4 + 0];
                avv.y = xb[j * 4 + 1];
                bvv.x = yb[j * 4 + 0];
                bvv.y = yb[j * 4 + 1];
                c = __builtin_amdgcn_wmma_f32_16x16x4_f32(
                        false, avv, false, bvv, (short)0, c, false, false);
            }
            // diagonal: lanes 0..7 -> diag[L] = c[L];
            //           lanes 24..31 -> diag[L-16] = c[L-24]
            if (L < 8)        Dpart[wv * TILE + L]        = c[L];
            else if (L >= 24) Dpart[wv * TILE + (L - 16)] = c[L - 24];
        }
        __syncthreads();

        // ---- phase 3: reduce 8 wave partials (fixed order), log ----
        if (tid < TILE) {
            const int li = t * TILE + tid;
            if (li < cnt) {
                float d = 0.f;
#pragma unroll
                for (int w = 0; w < NW; ++w) d += Dpart[w * TILE + tid];
                acc += __logf(mu_sh[li] + d);
            }
        }
        __syncthreads();   // protect Dpart/X/Y before next tile overwrites
    }
    if (tid < TILE) acc_sh[tid] = acc;
    __syncthreads();
    if (tid == 0) {
        float ssum = 0.f;
        for (int i = 0; i < TILE; ++i) ssum += acc_sh[i];   // fixed order
        partial[b] = ssum;
    }
}

// ============================================================
// Kernel 4: deterministic final reduction of block partials.
// ============================================================
__global__ void hawkes_k4_reduce(const float* __restrict__ partial,
                                 float* __restrict__ out, int B) {
    if (threadIdx.x == 0 && blockIdx.x == 0) {
        float s = 0.f;
        for (int i = 0; i < B; ++i) s += partial[i];
        out[0] = s;
    }
}

// ============================================================
extern "C" void kernel_launch(void* const* d_in, const int* in_sizes, int n_in,
                              void* d_out, int out_size, void* d_ws, size_t ws_size,
                              hipStream_t stream) {
    const float* ti    = (const float*)d_in[0];
    const int*   mi    = (const int*)  d_in[1];
    const float* mu    = (const float*)d_in[2];
    const float* alpha = (const float*)d_in[3];
    const float* gamma = (const float*)d_in[4];

    const int N = in_sizes[0];
    const int B = (N + CB - 1) / CB;

    // ws layout (floats): alphaT[50*200] | L[B*200] | A[B*4] | H[B*200] | partial[B]
    float* alphaT  = (float*)d_ws;
    float* Lw      = alphaT + (size_t)MM * MK;
    float* Aw      = Lw + (size_t)B * MK;
    float* Hw      = Aw + (size_t)B * KK;
    float* partial = Hw + (size_t)B * MK;

    hawkes_k0_alphaT<<<MM, NT, 0, stream>>>(alpha, gamma, alphaT);
    hawkes_k1_local <<<B, NT, 0, stream>>>(ti, mi, gamma, Lw, Aw, N);
    hawkes_k2_scan  <<<1, NT, 0, stream>>>(Lw, Aw, Hw, B);
    hawkes_k3_main  <<<B, NT, 0, stream>>>(ti, mi, mu, gamma, alphaT, Hw, partial, N);
    hawkes_k4_reduce<<<1, 64, 0, stream>>>(partial, (float*)d_out, B);
}